// InstSeg_18786186952796
// MI455X (gfx1250) — compile-verified
//
#include <hip/hip_runtime.h>
#include <hip/hip_bf16.h>

typedef float v2f __attribute__((ext_vector_type(2)));
typedef float v8f __attribute__((ext_vector_type(8)));

#define THRESH    0.1f
#define NMS_PAD   3
#define TOPK      200
#define KPAD      208            // 13 tiles of 16 centers
#define HDR_BYTES 8192
#define INVAL     1.0e10f

// ---------------------------------------------------------------------------
// K1: threshold + 7x7 max-pool NMS, compact surviving (score, idx) pairs.
// 16x16 output tile per block, 22x22 LDS halo tile (OOB = -inf, matching
// the reference's reduce_window padding).
// ---------------------------------------------------------------------------
__global__ void nms_compact_kernel(const float* __restrict__ hmp,
                                   float* __restrict__ cand_s,
                                   int* __restrict__ cand_i,
                                   int* __restrict__ cnt,
                                   int cap, int H, int W) {
    __shared__ float tile[22][22];
    const int bx = blockIdx.x * 16, by = blockIdx.y * 16;
    const int tid = threadIdx.y * 16 + threadIdx.x;
    for (int i = tid; i < 22 * 22; i += 256) {
        int ly = i / 22, lx = i % 22;
        int gy = by + ly - NMS_PAD, gx = bx + lx - NMS_PAD;
        float v = -__builtin_inff();
        if (gy >= 0 && gy < H && gx >= 0 && gx < W) {
            float h = hmp[gy * W + gx];
            v = (h > THRESH) ? h : -1.0f;
        }
        tile[ly][lx] = v;
    }
    __syncthreads();
    const int ty = threadIdx.y, tx = threadIdx.x;
    const float v = tile[ty + NMS_PAD][tx + NMS_PAD];
    float m = -__builtin_inff();
    #pragma unroll
    for (int dy = 0; dy < 7; ++dy)
        #pragma unroll
        for (int dx = 0; dx < 7; ++dx)
            m = fmaxf(m, tile[ty + dy][tx + dx]);
    if (v == m && v > 0.0f) {
        int pos = atomicAdd(cnt, 1);
        if (pos < cap) {
            cand_s[pos] = v;
            cand_i[pos] = (by + ty) * W + (bx + tx);
        }
    }
}

// ---------------------------------------------------------------------------
// K2: single-block top-200 selection (tie-break: lower flat index), stable
// rank-sort by row-major index among valid, emit centers/valid + augmented
// center matrix A[k] = (-2cy, -2cx, cy^2+cx^2, 0) with 1e10 sentinels.
// ---------------------------------------------------------------------------
__global__ void topk_select_kernel(float* __restrict__ cand_s,
                                   const int* __restrict__ cand_i,
                                   const int* __restrict__ cnt,
                                   int cap,
                                   float4* __restrict__ A,
                                   float* __restrict__ outC,   // [TOPK*2]
                                   float* __restrict__ outV,   // [TOPK]
                                   int H, int W) {
    const int tid = threadIdx.x;      // 256 threads
    __shared__ float rs[256];
    __shared__ int   ri[256];
    __shared__ int   rp[256];
    __shared__ float sel_s[TOPK];
    __shared__ int   sel_i[TOPK];
    __shared__ int   srt_i[TOPK];
    __shared__ int   srt_v[TOPK];
    int n = *cnt;
    if (n > cap) n = cap;
    const int HW = H * W;

    for (int r = 0; r < TOPK; ++r) {
        float bs = -3.0f; int bi = 0x7FFFFFFF; int bp = -1;
        for (int j = tid; j < n; j += 256) {
            float s = cand_s[j];
            if (s < -1.5f) continue;              // already selected
            int ix = cand_i[j];
            if (s > bs || (s == bs && ix < bi)) { bs = s; bi = ix; bp = j; }
        }
        rs[tid] = bs; ri[tid] = bi; rp[tid] = bp;
        __syncthreads();
        for (int off = 128; off > 0; off >>= 1) {
            if (tid < off) {
                float s2 = rs[tid + off]; int i2 = ri[tid + off];
                if (s2 > rs[tid] || (s2 == rs[tid] && i2 < ri[tid])) {
                    rs[tid] = s2; ri[tid] = i2; rp[tid] = rp[tid + off];
                }
            }
            __syncthreads();
        }
        if (tid == 0) {
            if (rp[0] >= 0) {
                sel_s[r] = rs[0]; sel_i[r] = ri[0];
                cand_s[rp[0]] = -2.0f;            // exclude from later rounds
            } else {
                sel_s[r] = -1.0f; sel_i[r] = HW;  // not enough candidates
            }
        }
        __syncthreads();
    }

    // stable rank sort by key = valid ? idx : HW  (matches reference order_key)
    if (tid < TOPK) {
        const int vi = sel_s[tid] > 0.0f;
        const int key = vi ? sel_i[tid] : HW;
        int rank = 0;
        for (int j = 0; j < TOPK; ++j) {
            const int vj = sel_s[j] > 0.0f;
            const int kj = vj ? sel_i[j] : HW;
            rank += (kj < key) || (kj == key && j < tid);
        }
        srt_i[rank] = sel_i[tid];
        srt_v[rank] = vi;
    }
    __syncthreads();

    if (tid < KPAD) {
        float fy = INVAL, fx = INVAL;
        if (tid < TOPK) {
            const int ix = srt_i[tid];
            const int cy = ix / W, cx = ix - (ix / W) * W;
            outC[2 * tid]     = (float)cy;
            outC[2 * tid + 1] = (float)cx;
            outV[tid] = srt_v[tid] ? 1.0f : 0.0f;
            if (srt_v[tid]) { fy = (float)cy; fx = (float)cx; }
        }
        A[tid] = make_float4(-2.0f * fy, -2.0f * fx, fy * fy + fx * fx, 0.0f);
    }
}

// ---------------------------------------------------------------------------
// K3: WMMA nearest-center assignment.
//   score[k,p] = (-2cy_k)*ly_p + (-2cx_k)*lx_p + (cy_k^2+cx_k^2)*1
// Each wave owns 16 pixels; B(4x16) = pixel features, A(16x4) = 16 centers.
// 13 V_WMMA_F32_16X16X4_F32 per wave cover KPAD=208 centers; argmin across
// accumulator rows + shfl_xor(16) combine of the two lane halves.
// ---------------------------------------------------------------------------
__global__ void assign_wmma_kernel(const float* __restrict__ offs,
                                   const int* __restrict__ sem,
                                   const float4* __restrict__ A,
                                   float* __restrict__ out,
                                   int H, int W) {
    __shared__ float4 sA[KPAD];
    const int tid = threadIdx.x;
    if (tid < KPAD) sA[tid] = A[tid];
    __syncthreads();

    const int HW = H * W;
    const int lane   = tid & 31;
    const int wave   = tid >> 5;
    const int lane16 = lane & 15;
    const bool hi    = lane >= 16;
    const int p = (blockIdx.x * 8 + wave) * 16 + lane16;
    const int y = p / W, x = p - y * W;
    const float ly = (float)y + offs[p];
    const float lx = (float)x + offs[HW + p];

    // B (4x16): lo half lanes carry rows K=0,1 = (ly, lx); hi half K=2,3 = (1, 0)
    v2f b;
    b[0] = hi ? 1.0f : ly;
    b[1] = hi ? 0.0f : lx;

    float bs = __builtin_inff();
    int   bk = 0;
    #pragma unroll
    for (int t = 0; t < KPAD / 16; ++t) {
        // A (16x4): lo half lanes carry K=0,1; hi half K=2,3 (ISA 7.12.2)
        const float4 av = sA[t * 16 + lane16];
        v2f a;
        a[0] = hi ? av.z : av.x;
        a[1] = hi ? av.w : av.y;
        v8f c = {0.f, 0.f, 0.f, 0.f, 0.f, 0.f, 0.f, 0.f};
        c = __builtin_amdgcn_wmma_f32_16x16x4_f32(
                false, a, false, b, (short)0, c, false, false);
        #pragma unroll
        for (int r = 0; r < 8; ++r) {
            const int k = t * 16 + r + (hi ? 8 : 0);   // C/D row mapping
            const float s = c[r];
            if (s < bs) { bs = s; bk = k; }            // first-min tie-break
        }
    }
    // combine the two lane halves holding the same pixel (rows split 0-7/8-15)
    const float os = __shfl_xor(bs, 16);
    const int   ok = __shfl_xor(bk, 16);
    if (os < bs || (os == bs && ok < bk)) { bs = os; bk = ok; }

    if (!hi) {
        const int sv = sem[p];
        const bool thing = (sv >= 11) && (sv <= 18);   // THING_LIST is contiguous
        out[p] = thing ? (float)(bk + 1) : 0.0f;
    }
}

// ---------------------------------------------------------------------------
extern "C" void kernel_launch(void* const* d_in, const int* in_sizes, int n_in,
                              void* d_out, int out_size, void* d_ws, size_t ws_size,
                              hipStream_t stream) {
    const int HW = in_sizes[1];          // ctr_hmp element count = H*W
    const int W = 1024;
    const int H = HW / W;

    const int*   sem  = (const int*)d_in[0];
    const float* hmp  = (const float*)d_in[1];
    const float* offs = (const float*)d_in[2];
    float* out = (float*)d_out;

    char* ws = (char*)d_ws;
    int*    cnt = (int*)ws;
    float4* A   = (float4*)(ws + 16);
    size_t avail = (ws_size > HDR_BYTES) ? (ws_size - HDR_BYTES) : 0;
    long long cap64 = (long long)(avail / 8);
    int cap = (cap64 > (long long)HW) ? HW : (int)cap64;
    float* cand_s = (float*)(ws + HDR_BYTES);
    int*   cand_i = (int*)(ws + HDR_BYTES + (size_t)cap * 4);

    hipMemsetAsync(cnt, 0, 16, stream);

    dim3 b1(16, 16);
    dim3 g1(W / 16, H / 16);
    nms_compact_kernel<<<g1, b1, 0, stream>>>(hmp, cand_s, cand_i, cnt, cap, H, W);

    topk_select_kernel<<<1, 256, 0, stream>>>(cand_s, cand_i, cnt, cap, A,
                                              out + HW, out + HW + 2 * TOPK, H, W);

    const int blocks = (HW / 16) / 8;    // 8 waves/block, 16 pixels/wave
    assign_wmma_kernel<<<blocks, 256, 0, stream>>>(offs, sem, A, out, H, W);
}